// GramAnchoring_67224828117335
// MI455X (gfx1250) — compile-verified
//
#include <hip/hip_runtime.h>

typedef float v2f __attribute__((ext_vector_type(2)));
typedef float v8f __attribute__((ext_vector_type(8)));
typedef unsigned int v4u __attribute__((ext_vector_type(4)));
typedef int v8i __attribute__((ext_vector_type(8)));
typedef int v4i __attribute__((ext_vector_type(4)));

#define CH   384      // channels
#define NBLK 256      // gram blocks per scale
#define TPB  384      // 12 waves of 32
#define NW   12
#define TS   33       // padded tile row stride (bank-conflict-free column reads)
#define TILEW (32 * TS)

__device__ __forceinline__ v8f wmma_16x16x4(v2f a, v2f b, v8f c) {
    return __builtin_amdgcn_wmma_f32_16x16x4_f32(false, a, false, b, (short)0, c, false, false);
}

// TDM: DMA a 32x32 f32 tile (row stride CH elements) from global into LDS,
// inserting 1 DWORD of padding every 32 DWORDs => LDS row stride = 33 words.
__device__ __forceinline__ void tdm_load_tile(const float* gbase, unsigned byteOff,
                                              unsigned ldsOff) {
    unsigned off = (unsigned)__builtin_amdgcn_readfirstlane((int)byteOff);
    unsigned lds = (unsigned)__builtin_amdgcn_readfirstlane((int)ldsOff);
    unsigned long long ga = (unsigned long long)(uintptr_t)gbase + (unsigned long long)off;

    v4u g0;
    g0.x = 1u;                                         // count=1 (valid), user descriptor
    g0.y = lds;                                        // lds_addr (bytes)
    g0.z = (unsigned)(ga & 0xFFFFFFFFull);             // global_addr[31:0]
    g0.w = (unsigned)((ga >> 32) & 0x01FFFFFFull)      // global_addr[56:32]
           | (2u << 30);                               // type = 2 ("image")

    v8i g1;
    g1[0] = (int)((2u << 16)                           // data_size = 4 bytes
                | (1u << 20)                           // pad_enable
                | (4u << 22));                         // pad_interval: code 4 = 32 DWORDs
                                                       // pad_amount:   code 0 = 1 DWORD
    g1[1] = 0;                                         // abar addr = 0; tensor_dim0 lo16 = 0
    g1[2] = 0x4000;                                    // tensor_dim0 = 1<<30 (never OOB)
    g1[3] = (int)((32u << 16) | 0x4000u);              // tile_dim0 = 32; tensor_dim1 = 1<<30
    g1[4] = 32;                                        // tile_dim1 = 32 (tile_dim2 = 0)
    g1[5] = CH;                                        // tensor_dim0_stride = 384 elements
    g1[6] = 0;
    g1[7] = 0;

    v4i gz4 = {0, 0, 0, 0};                            // groups 2/3: dims beyond 2D unused
    v8i gz8 = {0, 0, 0, 0, 0, 0, 0, 0};                // trailing group (clang-23 form): zero
    __builtin_amdgcn_tensor_load_to_lds(g0, g1, gz4, gz4, gz8, 0);
}

__global__ __launch_bounds__(TPB) void gram_main(const float* __restrict__ feats,
                                                 const float* __restrict__ anchors,
                                                 float* __restrict__ out,
                                                 float* __restrict__ lossWs)
{
    __shared__ float blkT[4 * CH];         // blkT[i][c] = blk[c][i]
    __shared__ float YT[4 * CH];           // Y = A   @ blk   (transposed [i][c])
    __shared__ float ZT[4 * CH];           // Z = A^T @ blk
    __shared__ float BST[4 * CH];          // blk @ S
    __shared__ float Sm[16];               // S = blk^T blk (4x4)
    __shared__ float tiles[2 * NW * TILEW];// double-buffered 32x32 tiles (stride 33)
    __shared__ float red[TPB];

    const int wg   = blockIdx.x;
    const int s    = wg >> 8;
    const int n    = wg & 255;
    const int bh   = n >> 4, bw = n & 15;
    const int tid  = threadIdx.x;
    const int w    = tid >> 5;
    const int lane = tid & 31;

    const long featBase = (long)s * 1025 * CH;

    // ---- load blk: row r_i = 1 + (bh*2+py)*32 + bw*2+px,  i = py*2+px ----
    #pragma unroll
    for (int i = 0; i < 4; ++i) {
        int py = i >> 1, px = i & 1;
        int row = 1 + (bh * 2 + py) * 32 + (bw * 2 + px);
        blkT[i * CH + tid] = feats[featBase + (long)row * CH + tid];
    }

    const float invcp2 = 1.0f / (float)(CH * 4);      // 1/1536
    const long  anchBase = ((long)(s * NBLK + n)) * CH * CH;

    // prologue: TDM round-0 tile (ti = w, tj = w) into buffer 0
    {
        unsigned bo = (unsigned)((anchBase + (long)(32 * w) * CH + 32 * w) * 4);
        tdm_load_tile(anchors, bo, (unsigned)(uintptr_t)&tiles[w * TILEW]);
    }

    __syncthreads();   // blkT visible

    // ---- S = blk^T blk, 16 threads ----
    if (tid < 16) {
        int i = tid & 3, j = tid >> 2;
        float acc = 0.f;
        for (int c = 0; c < CH; ++c) acc += blkT[i * CH + c] * blkT[j * CH + c];
        Sm[j * 4 + i] = acc;
    }

    float y0 = 0, y1 = 0, y2 = 0, y3 = 0;
    float z0 = 0, z1 = 0, z2 = 0, z3 = 0;
    float lossAcc = 0.f;

    const int m16  = lane & 15;
    const int k0   = (lane < 16) ? 0 : 2;      // K pair held by this lane half
    const int mAdd = (lane < 16) ? 0 : 8;      // D-matrix row offset for hi lanes

    for (int k = 0; k < NW; ++k) {
        const int buf   = k & 1;
        const int tiOwn = (w + k) % NW;        // row-tile of this wave's round-k tile
        __syncthreads();                       // round k-1 consumption finished everywhere
        if (k + 1 < NW) {
            // prefetch round k+1 via TDM into the other buffer (overlaps consumption)
            int tiN = (w + k + 1) % NW;
            unsigned bo = (unsigned)((anchBase + (long)(32 * tiN) * CH + 32 * w) * 4);
            tdm_load_tile(anchors, bo,
                          (unsigned)(uintptr_t)&tiles[((buf ^ 1) * NW + w) * TILEW]);
            __builtin_amdgcn_s_wait_tensorcnt(1);  // round k's TDM complete (in-order)
        } else {
            __builtin_amdgcn_s_wait_tensorcnt(0);
        }
        __syncthreads();                       // all 12 round-k tiles visible to all waves

        const float* tbuf   = &tiles[buf * NW * TILEW];
        const float* myTile = tbuf + w * TILEW;

        // Z += A^T blk over own columns (d = 32w+lane), rows c = 32*tiOwn+cl
        #pragma unroll 4
        for (int cl = 0; cl < 32; ++cl) {
            float a = myTile[cl * TS + lane];
            int cg = 32 * tiOwn + cl;
            z0 += a * blkT[0 * CH + cg];
            z1 += a * blkT[1 * CH + cg];
            z2 += a * blkT[2 * CH + cg];
            z3 += a * blkT[3 * CH + cg];
        }

        // Y += A blk over own rows (c = 32w+lane) using the tile with ti == w
        {
            int wo = (w - k + NW) % NW;        // staged by wave wo (its tj = wo)
            const float* oT = tbuf + wo * TILEW;
            #pragma unroll 4
            for (int dl = 0; dl < 32; ++dl) {
                float a = oT[lane * TS + dl];
                int dg = 32 * wo + dl;
                y0 += a * blkT[0 * CH + dg];
                y1 += a * blkT[1 * CH + dg];
                y2 += a * blkT[2 * CH + dg];
                y3 += a * blkT[3 * CH + dg];
            }
        }

        // loss: gram tiles via V_WMMA_F32_16X16X4_F32, diff vs staged anchor tile
        #pragma unroll
        for (int si = 0; si < 2; ++si) {
            int cbase = 32 * tiOwn + 16 * si;
            v2f af;
            af.x = blkT[k0 * CH + cbase + m16];
            af.y = blkT[(k0 + 1) * CH + cbase + m16];
            #pragma unroll
            for (int sj = 0; sj < 2; ++sj) {
                int dbase = 32 * w + 16 * sj;
                v2f bf;
                bf.x = blkT[k0 * CH + dbase + m16];
                bf.y = blkT[(k0 + 1) * CH + dbase + m16];
                v8f cz = {0, 0, 0, 0, 0, 0, 0, 0};
                v8f d = wmma_16x16x4(af, bf, cz);
                #pragma unroll
                for (int v = 0; v < 8; ++v) {
                    float g = d[v] * invcp2;
                    float a = myTile[(16 * si + v + mAdd) * TS + 16 * sj + m16];
                    float df = g - a;
                    lossAcc += df * df;
                }
            }
        }
    }

    // exclusive stores of per-wave strips
    YT[0 * CH + 32 * w + lane] = y0;
    YT[1 * CH + 32 * w + lane] = y1;
    YT[2 * CH + 32 * w + lane] = y2;
    YT[3 * CH + 32 * w + lane] = y3;
    ZT[0 * CH + 32 * w + lane] = z0;
    ZT[1 * CH + 32 * w + lane] = z1;
    ZT[2 * CH + 32 * w + lane] = z2;
    ZT[3 * CH + 32 * w + lane] = z3;

    // BS = blk @ S via WMMA (S padded 4x16); 24 row-fragments, 2 per wave
    {
        v2f bs;
        bs.x = (m16 < 4) ? Sm[k0 * 4 + m16] : 0.f;
        bs.y = (m16 < 4) ? Sm[(k0 + 1) * 4 + m16] : 0.f;
        #pragma unroll
        for (int rep = 0; rep < 2; ++rep) {
            int cbase = 16 * (w + rep * NW);
            v2f af;
            af.x = blkT[k0 * CH + cbase + m16];
            af.y = blkT[(k0 + 1) * CH + cbase + m16];
            v8f cz = {0, 0, 0, 0, 0, 0, 0, 0};
            v8f d = wmma_16x16x4(af, bs, cz);
            if (m16 < 4) {
                #pragma unroll
                for (int v = 0; v < 8; ++v)
                    BST[m16 * CH + cbase + v + mAdd] = d[v];
            }
        }
    }
    __syncthreads();

    // grad writeback: grad = kgrad*(2*invcp2*BS - Y - Z); enhanced = feats + 0.01*grad
    const float kgrad = invcp2 * 2.0f / ((float)NBLK * (float)CH * (float)CH);
    #pragma unroll
    for (int i = 0; i < 4; ++i) {
        int py = i >> 1, px = i & 1;
        int row = 1 + (bh * 2 + py) * 32 + (bw * 2 + px);
        float grad = kgrad * (2.f * invcp2 * BST[i * CH + tid]
                              - YT[i * CH + tid] - ZT[i * CH + tid]);
        long oIdx = featBase + (long)row * CH + tid;
        out[oIdx] = feats[oIdx] + 0.01f * grad;
    }
    if (n == 0) {  // CLS token row: gradient is zero
        out[featBase + tid] = feats[featBase + tid];
    }

    // deterministic loss reduction -> workspace
    red[tid] = lossAcc;
    __syncthreads();
    if (tid < 128) red[tid] = red[tid] + red[tid + 128] + red[tid + 256];
    __syncthreads();
    for (int off = 64; off > 0; off >>= 1) {
        if (tid < off) red[tid] += red[tid + off];
        __syncthreads();
    }
    if (tid == 0)
        lossWs[wg] = red[0] / ((float)NBLK * (float)CH * (float)CH);
}

__global__ void loss_reduce(const float* __restrict__ ws, float* __restrict__ out,
                            int nWg, long outIdx)
{
    int lane = threadIdx.x;
    float acc = 0.f;
    for (int j = lane; j < nWg; j += 32) acc += ws[j];
    #pragma unroll
    for (int off = 16; off > 0; off >>= 1) acc += __shfl_down(acc, off, 32);
    if (lane == 0) out[outIdx] = acc;
}

extern "C" void kernel_launch(void* const* d_in, const int* in_sizes, int n_in,
                              void* d_out, int out_size, void* d_ws, size_t ws_size,
                              hipStream_t stream) {
    const float* feats   = (const float*)d_in[0];   // [4,1,1025,384]
    const float* anchors = (const float*)d_in[1];   // [4,1,256,384,384]
    float* out = (float*)d_out;                     // [4*1025*384] enhanced + [1] loss
    float* ws  = (float*)d_ws;                      // 1024 loss partials

    gram_main<<<dim3(4 * NBLK), dim3(TPB), 0, stream>>>(feats, anchors, out, ws);
    loss_reduce<<<dim3(1), dim3(32), 0, stream>>>(ws, out, 1024, (long)4 * 1025 * 384);
}